// LayerBase_29841432773434
// MI455X (gfx1250) — compile-verified
//
#include <hip/hip_runtime.h>
#include <cstdint>

// Problem dims (fixed by the reference)
#define BDIM 16
#define NDIM 8192
#define MDIM 2048
#define KDIM 32
#define CDIM 128

typedef float v4f __attribute__((ext_vector_type(4)));

static_assert(MDIM * KDIM == 65536, "b = row >> 16 relies on M*K == 2^16");

#if __has_builtin(__builtin_amdgcn_s_wait_asynccnt)
#define WAIT_ASYNC(n) __builtin_amdgcn_s_wait_asynccnt(n)
#else
#define WAIT_ASYNC(n) asm volatile("s_wait_asynccnt %0" ::"i"(n) : "memory")
#endif

// ---------------------------------------------------------------------------
// Kernel 1: pts = (points[b,idx] - next_pts[b,m]) / maxnorm over K neighbors
// One wave per (b,m); lane == neighbor k (K == 32 == wave32).
// ---------------------------------------------------------------------------
__global__ void __launch_bounds__(256) pts_kernel(
    const float* __restrict__ points,    // [B,N,3]
    const float* __restrict__ next_pts,  // [B,M,3]
    const int* __restrict__ indices,     // [B,M,K]
    float* __restrict__ out_pts)         // [B,M,K,3]
{
  const unsigned tid  = blockIdx.x * blockDim.x + threadIdx.x;
  const unsigned wave = tid >> 5;   // (b*M + m)
  const unsigned lane = tid & 31u;  // k
  if (wave >= (unsigned)(BDIM * MDIM)) return;
  const unsigned b = wave / MDIM;

  const int idx = indices[(size_t)wave * KDIM + lane];
  const float* p = points + ((size_t)b * NDIM + (unsigned)idx) * 3;
  const float* q = next_pts + (size_t)wave * 3;

  const float dx = p[0] - q[0];
  const float dy = p[1] - q[1];
  const float dz = p[2] - q[2];
  const float sq = dx * dx + dy * dy + dz * dz;

  // wave32 max reduction over the K=32 neighbors
  float mx = sq;
#pragma unroll
  for (int off = 16; off > 0; off >>= 1)
    mx = fmaxf(mx, __shfl_xor(mx, off, 32));

  float nrm = sqrtf(mx);
  if (nrm == 0.0f) nrm = 1.0f;

  float* o = out_pts + ((size_t)wave * KDIM + lane) * 3;
  o[0] = dx / nrm;
  o[1] = dy / nrm;
  o[2] = dz / nrm;
}

// ---------------------------------------------------------------------------
// Kernel 2: features[b,m,k,:] = input[b, indices[b,m,k], :]
// One wave handles 32 consecutive rows (512B each). All 32 neighbor indices
// are fetched with ONE coalesced load, then broadcast per iteration via
// readlane. Gathers go global->LDS through a 6-deep async ring (ASYNCcnt),
// drained with non-temporal 128-bit stores so the 537MB output stream does
// not evict the 64MB feature table from the 192MB L2.
// The 32-step pipeline is template-unrolled so every s_wait_asynccnt
// argument is a frontend integer constant (builtin requirement).
// ---------------------------------------------------------------------------
#define FEAT_BLOCKS 4096
#define WPB 8            // waves per 256-thread block
#define ROWS_PER_WAVE 32
#define DEPTH 6          // async ring depth (buffers in flight <= DEPTH-1)

struct FeatCtx {
  const v4f* __restrict__ inb;  // batch base of input  [N, C/4]
  v4f* __restrict__ out4;       // output               [ROWS, C/4]
  v4f* shbase;                  // &sh[wib][0][0] (this wave's LDS ring)
  const uint32_t* ldsoff;       // DEPTH per-lane LDS byte offsets
  int idxs;                     // lane l holds neighbor index for step l
  unsigned lane;
  unsigned base;                // first row of this wave
};

__device__ __forceinline__ void issue_row(const FeatCtx& c, int i) {
  const unsigned idx = (unsigned)__shfl(c.idxs, i, 32);  // readlane broadcast
  const v4f* src = c.inb + (size_t)idx * (CDIM / 4) + c.lane;
  // Per-lane 16B -> whole wave moves one 512B feature row into LDS.
  asm volatile("global_load_async_to_lds_b128 %0, %1, off"
               :: "v"(c.ldsoff[i % DEPTH]), "v"(src)
               : "memory");
}

template <int I>
__device__ __forceinline__ void feat_steps(const FeatCtx& c) {
  if constexpr (I < ROWS_PER_WAVE) {
    constexpr int SS  = DEPTH - 2;              // steady-state outstanding-1
    constexpr int REM = ROWS_PER_WAVE - 1 - I;  // loads still outstanding-1
    WAIT_ASYNC((REM < SS) ? REM : SS);          // oldest slot now resident

    const v4f v = c.shbase[(I % DEPTH) * 32 + c.lane];
    __builtin_nontemporal_store(
        v, c.out4 + (size_t)(c.base + I) * (CDIM / 4) + c.lane);

    // Refill recycled slot AFTER reading it (WAR-safe).
    if constexpr (I + DEPTH - 1 < ROWS_PER_WAVE) issue_row(c, I + DEPTH - 1);

    feat_steps<I + 1>(c);
  }
}

__global__ void __launch_bounds__(256) feat_kernel(
    const v4f* __restrict__ input4,   // [B,N,C/4]
    const int* __restrict__ indices,  // flat [B*M*K]
    v4f* __restrict__ out4)           // [B*M*K, C/4]
{
  __shared__ v4f sh[WPB][DEPTH][32];

  const unsigned lane = threadIdx.x & 31u;
  const unsigned wib  = threadIdx.x >> 5;
  const unsigned wave = blockIdx.x * WPB + wib;

  constexpr unsigned ROWS = (unsigned)BDIM * MDIM * KDIM;  // 2^20
  static_assert(ROWS == (unsigned)FEAT_BLOCKS * WPB * ROWS_PER_WAVE,
                "exact tiling");

  const unsigned base = wave * ROWS_PER_WAVE;  // first row of this wave
  // 32 consecutive rows never cross a batch boundary (65536 % 32 == 0)
  const unsigned b = base >> 16;

  uint32_t ldsoff[DEPTH];
#pragma unroll
  for (int d = 0; d < DEPTH; ++d)
    ldsoff[d] = (uint32_t)(uintptr_t)&sh[wib][d][lane];

  FeatCtx c;
  c.inb    = input4 + (size_t)b * (NDIM * (CDIM / 4));
  c.out4   = out4;
  c.shbase = &sh[wib][0][0];
  c.ldsoff = ldsoff;
  c.idxs   = indices[base + lane];  // ONE coalesced 128B index load
  c.lane   = lane;
  c.base   = base;

  // prologue: fill DEPTH-1 ring slots
#pragma unroll
  for (int d = 0; d < DEPTH - 1; ++d) issue_row(c, d);

  // template-unrolled steady state + tail
  feat_steps<0>(c);
}

// ---------------------------------------------------------------------------
extern "C" void kernel_launch(void* const* d_in, const int* in_sizes, int n_in,
                              void* d_out, int out_size, void* d_ws,
                              size_t ws_size, hipStream_t stream)
{
  const float* input    = (const float*)d_in[0];  // [B,N,C]
  const float* points   = (const float*)d_in[1];  // [B,N,3]
  const float* next_pts = (const float*)d_in[2];  // [B,M,3]
  const int*   indices  = (const int*)d_in[3];    // [B,M,K]

  float* out_pts  = (float*)d_out;                             // [B,M,K,3]
  float* out_feat = out_pts + (size_t)BDIM * MDIM * KDIM * 3;  // [B,M,K,C]

  pts_kernel<<<(BDIM * MDIM) / WPB, 256, 0, stream>>>(points, next_pts,
                                                      indices, out_pts);
  feat_kernel<<<FEAT_BLOCKS, 256, 0, stream>>>(
      (const v4f*)input, indices, (v4f*)out_feat);
}